// YoloV1Loss_37280316129412
// MI455X (gfx1250) — compile-verified
//
#include <hip/hip_runtime.h>
#include <cstdint>

// Problem constants (fixed by the reference): S=7, B=2, C=20, T=16
#define ROW       30          // B*5 + C
#define IMG_DW    1470        // 7*7*30
#define IMG_BYTES 5880        // IMG_DW * 4
#define SLAB_DW   1472        // padded slab stride (5888 B, 8-byte multiple)
#define WPB       8           // waves per block (one image per wave)
#define NTGT      16
#define NCLS      20
#define CHUNKS64  735         // IMG_BYTES / 8

// Tensor Data Mover builtin: 5 args on ROCm 7.2 (clang-22), 6 args on the
// clang-23 toolchain. Dispatch on availability + clang major; fall back to
// the (also CDNA5-specific) async global->LDS instruction path otherwise.
#if __has_builtin(__builtin_amdgcn_tensor_load_to_lds)
#define USE_TDM 1
#else
#define USE_TDM 0
#endif

typedef unsigned int u32x4 __attribute__((ext_vector_type(4)));
typedef int          i32x4 __attribute__((ext_vector_type(4)));
typedef int          i32x8 __attribute__((ext_vector_type(8)));

__device__ __forceinline__ uint32_t lds_addr32(const void* p) {
    // generic LDS pointer: low 32 bits == LDS byte offset (aperture in high bits)
    return (uint32_t)(uintptr_t)p;
}

__global__ __launch_bounds__(256) void yolo_loss_stage1(
    const float* __restrict__ outputs,
    const float* __restrict__ boxes,
    const int*   __restrict__ labels,
    float* __restrict__ block_partials)
{
    __shared__ float    slab[WPB * SLAB_DW];
    __shared__ unsigned maskw[WPB * 4];
    __shared__ float    wsum[WPB];

    const int lane = threadIdx.x & 31;
    const int wave = threadIdx.x >> 5;
    const int n    = blockIdx.x * WPB + wave;   // image index (N divisible by WPB)

    if (lane < 4) maskw[wave * 4 + lane] = 0u;

    const char*    gbase    = (const char*)outputs + (size_t)n * IMG_BYTES;
    const uint32_t lds_base = lds_addr32(&slab[wave * SLAB_DW]);

#if USE_TDM
    // ---- CDNA5 Tensor Data Mover: one DMA descriptor per image slab ----
    {
        const uint64_t ga = (uint64_t)(uintptr_t)gbase;
        // All fields are wave-uniform; force into SGPRs for the descriptor.
        const unsigned w0 = 1u;                                   // count=1, user desc
        const unsigned w1 = (unsigned)__builtin_amdgcn_readfirstlane((int)lds_base);
        const unsigned w2 = (unsigned)__builtin_amdgcn_readfirstlane((int)(uint32_t)ga);
        const unsigned w3 = (unsigned)__builtin_amdgcn_readfirstlane(
            (int)((((uint32_t)(ga >> 32)) & 0x01FFFFFFu) | 0x80000000u)); // type=2
        u32x4 g0 = {w0, w1, w2, w3};

        // Group 1: 1-D tensor of 1470 dword elements, tile == tensor.
        i32x8 g1 = {
            0x00020000,            // data_size=2 (4B); mask/pad/iterate/barrier = 0
            (int)(1470u << 16),    // tensor_dim0[15:0] in bits 63:48
            0x00010000,            // tensor_dim0[31:16]=0, tensor_dim1=1
            (int)(1470u << 16),    // tensor_dim1[31:16]=0, tile_dim0=1470
            1,                     // tile_dim1=1, tile_dim2=0
            1470,                  // tensor_dim0_stride[31:0]
            0,                     // stride0[47:32]=0, stride1[15:0]=0
            0                      // stride1[47:16]=0
        };
        i32x4 gz4 = {0, 0, 0, 0};  // groups 2/3: 2-D descriptor, higher dims unused
#if defined(__clang_major__) && (__clang_major__ <= 22)
        __builtin_amdgcn_tensor_load_to_lds(g0, g1, gz4, gz4, 0);
#else
        i32x8 gz8 = {0, 0, 0, 0, 0, 0, 0, 0};
        __builtin_amdgcn_tensor_load_to_lds(g0, g1, gz4, gz4, gz8, 0);
#endif
    }
#else
    // ---- CDNA5 async global->LDS staging of this image's slab ----
#pragma unroll
    for (int i = 0; i < 23; ++i) {
        int chunk = i * 32 + lane;
        if (chunk < CHUNKS64) {
            uint32_t    l = lds_base + chunk * 8;
            const char* g = gbase + chunk * 8;
            asm volatile("global_load_async_to_lds_b64 %0, %1, off"
                         :: "v"(l), "v"(g) : "memory");
        }
    }
#endif

    // ---- overlap: per-target scalar loads from global ----
    float tx = 0.f, ty = 0.f, tw = 0.f, th = 0.f;
    int   lab = 1, gxi = 0, gyi = 0;
    float offx = 0.f, offy = 0.f;
    if (lane < NTGT) {
        const float* bp = boxes + ((size_t)n * NTGT + lane) * 4;
        tx = bp[0]; ty = bp[1]; tw = bp[2]; th = bp[3];
        lab = labels[(size_t)n * NTGT + lane];
        float txs = tx * 7.0f, tys = ty * 7.0f;
        gxi = (int)txs; gyi = (int)tys;
        offx = txs - (float)gxi;
        offy = tys - (float)gyi;
    }

    // wait for this wave's DMA / async copies to land in LDS
#if USE_TDM
    __builtin_amdgcn_s_wait_tensorcnt(0);
#else
    asm volatile("s_wait_asynccnt 0" ::: "memory");
#endif
    __syncthreads();   // orders mask init vs atomicOr (uniform, cheap)

    const float* sm = &slab[wave * SLAB_DW];
    float part = 0.0f;

    if (lane < NTGT) {
        const int cbase = (gyi * 7 + gxi) * ROW;
        float iou[2], pxa[2], pya[2], pw[2], ph[2], cf[2];
#pragma unroll
        for (int b = 0; b < 2; ++b) {
            const float* q = sm + cbase + b * 5;
            float bx = q[0], by = q[1], bw = q[2], bh = q[3];
            cf[b] = q[4];
            float px = (bx + (float)gxi) * (1.0f / 7.0f);
            float py = (by + (float)gyi) * (1.0f / 7.0f);
            pxa[b] = px; pya[b] = py; pw[b] = bw; ph[b] = bh;
            // IoU(target, pred)
            float x1a = tx - tw * 0.5f, y1a = ty - th * 0.5f;
            float x2a = tx + tw * 0.5f, y2a = ty + th * 0.5f;
            float x1b = px - bw * 0.5f, y1b = py - bh * 0.5f;
            float x2b = px + bw * 0.5f, y2b = py + bh * 0.5f;
            float iw = fmaxf(fminf(x2a, x2b) - fmaxf(x1a, x1b), 0.0f);
            float ih = fmaxf(fminf(y2a, y2b) - fmaxf(y1a, y1b), 0.0f);
            float inter = iw * ih;
            float uni   = tw * th + bw * bh - inter;
            iou[b] = inter / fmaxf(uni, 1e-10f);
        }
        // reference tie rule: argmax over reversed axis -> ties pick b=1
        int best = (iou[1] >= iou[0]) ? 1 : 0;

        const float eps = 1e-6f;
        float d0 = pxa[best] - offx;
        float d1 = pya[best] - offy;
        float d2 = sqrtf(fmaxf(pw[best], eps)) - sqrtf(fmaxf(tw, eps));
        float d3 = sqrtf(fmaxf(ph[best], eps)) - sqrtf(fmaxf(th, eps));
        float box_loss = d0 * d0 + d1 * d1 + d2 * d2 + d3 * d3;
        float dc = cf[best] - 1.0f;
        float obj_loss = dc * dc;

        // softmax class loss over 20 logits (from LDS)
        const float* cl = sm + cbase + 10;
        float m = cl[0];
#pragma unroll
        for (int c = 1; c < NCLS; ++c) m = fmaxf(m, cl[c]);
        float e[NCLS];
        float esum = 0.0f;
#pragma unroll
        for (int c = 0; c < NCLS; ++c) { float ee = expf(cl[c] - m); e[c] = ee; esum += ee; }
        float inv = 1.0f / esum;
        float closs = 0.0f;
#pragma unroll
        for (int c = 0; c < NCLS; ++c) {
            float p = e[c] * inv - ((c == lab - 1) ? 1.0f : 0.0f);
            closs += p * p;
        }

        int slot = (gyi * 7 + gxi) * 2 + best;        // [0, 98)
        atomicOr(&maskw[wave * 4 + (slot >> 5)], 1u << (slot & 31));

        part = 5.0f * box_loss + obj_loss + closs;    // LC = 5.0
    }
    __syncthreads();   // mask bits visible before the no-obj sweep

    // ---- no-object loss: sweep 98 conf values from LDS ----
    int pc = __popc(maskw[wave * 4 + 0]) + __popc(maskw[wave * 4 + 1]) +
             __popc(maskw[wave * 4 + 2]) + __popc(maskw[wave * 4 + 3]);
    float cnt = (float)(98 - pc);

    float nsum = 0.0f;
    for (int j = lane; j < 98; j += 32) {
        unsigned w   = maskw[wave * 4 + (j >> 5)];
        unsigned bit = (w >> (j & 31)) & 1u;
        float c = sm[(j >> 1) * ROW + (j & 1) * 5 + 4];
        if (!bit) nsum += c * c;
    }

    // ---- wave32 reduction ----
    float v = part, u = nsum;
#pragma unroll
    for (int off = 16; off > 0; off >>= 1) {
        v += __shfl_xor(v, off, 32);
        u += __shfl_xor(u, off, 32);
    }
    if (lane == 0) wsum[wave] = v + 0.5f * (u / cnt);  // LN = 0.5
    __syncthreads();

    if (threadIdx.x == 0) {
        float s = 0.0f;
#pragma unroll
        for (int w = 0; w < WPB; ++w) s += wsum[w];
        block_partials[blockIdx.x] = s;
    }
}

__global__ __launch_bounds__(256) void yolo_loss_stage2(
    const float* __restrict__ parts, float* __restrict__ out,
    int nparts, float invN)
{
    __shared__ float red[256];
    float s = 0.0f;
    for (int i = threadIdx.x; i < nparts; i += 256) s += parts[i];
    red[threadIdx.x] = s;
    __syncthreads();
#pragma unroll
    for (int stride = 128; stride > 0; stride >>= 1) {
        if ((int)threadIdx.x < stride) red[threadIdx.x] += red[threadIdx.x + stride];
        __syncthreads();
    }
    if (threadIdx.x == 0) out[0] = red[0] * invN;
}

extern "C" void kernel_launch(void* const* d_in, const int* in_sizes, int n_in,
                              void* d_out, int out_size, void* d_ws, size_t ws_size,
                              hipStream_t stream) {
    const float* outputs = (const float*)d_in[0];
    const float* boxes   = (const float*)d_in[1];
    const int*   labels  = (const int*)d_in[2];

    const int N = in_sizes[0] / IMG_DW;          // 16384 (divisible by WPB)
    const int blocks = N / WPB;                  // 2048

    float* partials = (float*)d_ws;              // blocks * 4 bytes of scratch

    yolo_loss_stage1<<<blocks, 256, 0, stream>>>(outputs, boxes, labels, partials);
    yolo_loss_stage2<<<1, 256, 0, stream>>>(partials, (float*)d_out, blocks,
                                            1.0f / (float)N);
}